// InteractionPruningEluLayerV3_72378788872998
// MI455X (gfx1250) — compile-verified
//
#include <hip/hip_runtime.h>
#include <hip/hip_bf16.h>
#include <math.h>

// B=512, A=3, N=64, D=32 ; fp32 in / fp32 out
#define B_ 512
#define N_ 64
#define D_ 32

typedef __attribute__((ext_vector_type(2))) float v2f;
typedef __attribute__((ext_vector_type(8))) float v8f;

// ---------------------------------------------------------------------------
// Kernel 1: gate[i][j] = 1[ ind0[i,:] . ind1[j,:] > 0 ]  -> d_ws (64*64 floats)
// Tiny (131K MACs); one block, scalar is fine. Runs once per launch.
// ---------------------------------------------------------------------------
__global__ __launch_bounds__(256) void gate_kernel(const float* __restrict__ ind,
                                                   float* __restrict__ gate) {
  const float* ind0 = ind;              // [64][32]
  const float* ind1 = ind + N_ * D_;    // [64][32]
  for (int p = threadIdx.x; p < N_ * N_; p += 256) {
    int i = p >> 6, j = p & 63;
    float acc = 0.f;
#pragma unroll
    for (int d = 0; d < D_; ++d) acc += ind0[i * D_ + d] * ind1[j * D_ + d];
    gate[p] = (acc > 0.f) ? 1.f : 0.f;
  }
}

// ---------------------------------------------------------------------------
// Kernel 2: one workgroup (256 thr = 8 wave32) per batch element b.
//  B: s[a][n] = feat[n,:] . ind[a][n,:]
//  C: score[i][j] = sum_d (elu(s0[i]*ind0[j,d])+1)*(elu(s1[j]*ind1[i,d])+1)
//     (elu(x)+1 == x+1 if x>0 else exp(x))  -- dominant VALU cost
//  D: row softmax * gate
//  E: out[i][d] = s2[i] * (score_hat @ ind2)[i][d]  via V_WMMA_F32_16X16X4_F32
//     8 output tiles (16x16), one per wave; K=64 -> 16 WMMA steps per tile.
// ---------------------------------------------------------------------------
__global__ __launch_bounds__(256) void fused_kernel(const float* __restrict__ feature,
                                                    const float* __restrict__ ind,
                                                    const float* __restrict__ gate_g,
                                                    float* __restrict__ out) {
  __shared__ float s_ind[3][N_][D_];     // 24 KB
  __shared__ float s_feat[N_][D_];       //  8 KB
  __shared__ float s_s[3][N_];           //  0.75 KB
  __shared__ float s_score[N_][N_ + 1];  // 16.6 KB (padded: conflict-free row reads)

  const int t = threadIdx.x;
  const int b = blockIdx.x;

  // --- stage inputs to LDS (vectorized, coalesced) ---
  {
    const float4* gsrc = (const float4*)ind;
    float4* ldst = (float4*)&s_ind[0][0][0];
    for (int i = t; i < (3 * N_ * D_) / 4; i += 256) ldst[i] = gsrc[i];
    const float4* fsrc = (const float4*)(feature + (size_t)b * N_ * D_);
    float4* fdst = (float4*)&s_feat[0][0];
    for (int i = t; i < (N_ * D_) / 4; i += 256) fdst[i] = fsrc[i];
  }
  __syncthreads();

  // --- step B: s[a][n] (192 dot-32s) ---
  if (t < 3 * N_) {
    int a = t >> 6, n = t & 63;
    float acc = 0.f;
#pragma unroll
    for (int d = 0; d < D_; ++d) acc += s_feat[n][d] * s_ind[a][n][d];
    s_s[a][n] = acc;
  }
  __syncthreads();

  // --- step C: score (elu-bound; 16 (i,j) pairs per thread) ---
  for (int p = t; p < N_ * N_; p += 256) {
    int i = p >> 6, j = p & 63;
    float s0 = s_s[0][i], s1 = s_s[1][j];
    float acc = 0.f;
#pragma unroll
    for (int d = 0; d < D_; ++d) {
      float t0 = s0 * s_ind[0][j][d];
      float q = (t0 > 0.f) ? (t0 + 1.f) : __expf(t0);   // elu(x)+1
      float t1 = s1 * s_ind[1][i][d];
      float k = (t1 > 0.f) ? (t1 + 1.f) : __expf(t1);
      acc += q * k;
    }
    s_score[i][j] = acc;
  }
  __syncthreads();

  // --- step D: row softmax * gate (gate stays in L2; only 4K reads/block) ---
  if (t < N_) {
    int i = t;
    float m = -INFINITY;
    for (int j = 0; j < N_; ++j) m = fmaxf(m, s_score[i][j]);
    float sum = 0.f;
    for (int j = 0; j < N_; ++j) {
      float e = __expf(s_score[i][j] - m);
      s_score[i][j] = e;
      sum += e;
    }
    float inv = 1.f / sum;
    for (int j = 0; j < N_; ++j) s_score[i][j] *= inv * gate_g[i * N_ + j];
  }
  __syncthreads();

  // --- step E: batched GEMM tile via fp32 WMMA ---
  // wave w owns tile (ti = w/2, td = w%2): rows ti*16..+15, cols td*16..+15
  const int wave = t >> 5;        // wave32
  const int lane = t & 31;
  const int ti = wave >> 1;
  const int td = wave & 1;
  const int mrow = lane & 15;             // A row / C col index within tile
  const int koff = (lane < 16) ? 0 : 2;   // K sub-offset for upper lane half

  v8f acc = {};
#pragma unroll
  for (int kb = 0; kb < 16; ++kb) {       // K = 64 in steps of 4
    const int k0 = kb * 4 + koff;
    v2f afrag, bfrag;
    // A = score_hat tile: lane -> row (ti*16+mrow), VGPR v -> K = k0+v
    afrag.x = s_score[ti * 16 + mrow][k0];
    afrag.y = s_score[ti * 16 + mrow][k0 + 1];
    // B = ind2 (V): VGPR v -> K-row (kb*4 + v + koff), col = td*16 + lane%16
    bfrag.x = s_ind[2][kb * 4 + 0 + koff][td * 16 + mrow];
    bfrag.y = s_ind[2][kb * 4 + 1 + koff][td * 16 + mrow];
    acc = __builtin_amdgcn_wmma_f32_16x16x4_f32(
        /*neg_a=*/false, afrag, /*neg_b=*/false, bfrag,
        /*c_mod=*/(short)0, acc, /*reuse_a=*/false, /*reuse_b=*/false);
  }

  // C/D layout: VGPR r -> row r (lanes 0-15) / r+8 (lanes 16-31); col = lane%16
  const int rbase = (lane < 16) ? 0 : 8;
  float* outb = out + (size_t)b * N_ * D_;
#pragma unroll
  for (int r = 0; r < 8; ++r) {
    int i = ti * 16 + rbase + r;
    int d = td * 16 + mrow;
    outb[i * D_ + d] = s_s[2][i] * acc[r];   // fold diag(s2) scaling
  }
}

// ---------------------------------------------------------------------------
extern "C" void kernel_launch(void* const* d_in, const int* in_sizes, int n_in,
                              void* d_out, int out_size, void* d_ws, size_t ws_size,
                              hipStream_t stream) {
  const float* feature = (const float*)d_in[0];    // [512, 64, 32] fp32
  const float* indicator = (const float*)d_in[1];  // [3, 64, 32] fp32
  float* outp = (float*)d_out;                     // [512, 64, 32] fp32
  float* gate = (float*)d_ws;                      // 64*64 floats scratch

  gate_kernel<<<1, 256, 0, stream>>>(indicator, gate);
  fused_kernel<<<B_, 256, 0, stream>>>(feature, indicator, gate, outp);
}